// DeepFilter_22230750724156
// MI455X (gfx1250) — compile-verified
//
#include <hip/hip_runtime.h>

typedef float v2f __attribute__((ext_vector_type(2)));
typedef float v8f __attribute__((ext_vector_type(8)));

#define B_ 8
#define D_ 256
#define T_ 2048
#define L_ 2
#define I_ 1

#define DT 16                 // output d-rows per block
#define TT 128                // output t-cols per block
#define HR (DT + 2 * I_)      // 18 halo rows
#define HC (TT + 2 * L_)      // 132 halo cols
#define PSTR (HC + 1)         // padded LDS stride for products
#define SSTR TT               // stride for time-summed buffer

__global__ __launch_bounds__(256)
void deepfilter_kernel(const float* __restrict__ xr,
                       const float* __restrict__ xi,
                       const float* __restrict__ fr,
                       const float* __restrict__ fi,
                       float* __restrict__ out) {
  __shared__ float PR[HR * PSTR];
  __shared__ float PI[HR * PSTR];
  __shared__ float SR[HR * SSTR];
  __shared__ float SI[HR * SSTR];

  const int t0  = blockIdx.x * TT;
  const int d0  = blockIdx.y * DT;
  const int b   = blockIdx.z;
  const int tid = threadIdx.x;

  // ---- Phase 1: pointwise "complex" products into LDS (zero-padded halo) ----
  for (int i = tid; i < HR * HC; i += 256) {
    const int r  = i / HC;
    const int c  = i - r * HC;
    const int dg = d0 - I_ + r;
    const int tg = t0 - L_ + c;
    float ar = 0.f, ai = 0.f, br = 0.f, bi = 0.f;
    if (dg >= 0 && dg < D_ && tg >= 0 && tg < T_) {
      const size_t idx = ((size_t)(b * D_ + dg)) * T_ + tg;
      ar = xr[idx]; ai = xi[idx]; br = fr[idx]; bi = fi[idx];
    }
    PR[r * PSTR + c] = ar * br - ai * bi;
    PI[r * PSTR + c] = 2.0f * ar * bi;   // faithful to reference bug
  }
  __syncthreads();

  // ---- Phase 2a: 5-tap time box-sum (VALU) ----
  for (int i = tid; i < HR * SSTR; i += 256) {   // 2304 = 9 full rounds
    const int r = i >> 7;
    const int t = i & (TT - 1);
    const float* pr = &PR[r * PSTR + t];
    const float* pi = &PI[r * PSTR + t];
    SR[r * SSTR + t] = pr[0] + pr[1] + pr[2] + pr[3] + pr[4];
    SI[r * SSTR + t] = pi[0] + pi[1] + pi[2] + pi[3] + pi[4];
  }
  __syncthreads();

  // ---- Phase 2b: 3-tap freq sum as banded GEMM on the WMMA pipe ----
  // Out[16 x 16] = W(16x20, band) x S(20x16); 5 K-chunks of 4, C-chained.
  // Wave w owns t-subtile [16w, 16w+16). EXEC all-1s here (uniform code).
  const int wave = tid >> 5;
  const int lane = tid & 31;
  const int half = lane >> 4;     // low/high lane halves split K
  const int nm   = lane & 15;     // M for A-operand, N for B/D-operands
  const int tsub = wave * 16;

  v8f accR = {};
  v8f accI = {};
#pragma unroll
  for (int kc = 0; kc < 5; ++kc) {
    const int kg0 = kc * 4 + half * 2;   // global K of this lane's a.x / b.x
    const int kg1 = kg0 + 1;
    v2f a, bR, bI;
    // band: out row m sums halo rows [m, m+2]  (d-1, d, d+1)
    a.x = (kg0 >= nm && kg0 <= nm + 2) ? 1.0f : 0.0f;
    a.y = (kg1 >= nm && kg1 <= nm + 2) ? 1.0f : 0.0f;
    // K = 18,19 are padding (weight 0); clamp read so 0 * finite = 0 (no NaN)
    const int r0 = kg0 < HR ? kg0 : HR - 1;
    const int r1 = kg1 < HR ? kg1 : HR - 1;
    bR.x = SR[r0 * SSTR + tsub + nm];
    bR.y = SR[r1 * SSTR + tsub + nm];
    bI.x = SI[r0 * SSTR + tsub + nm];
    bI.y = SI[r1 * SSTR + tsub + nm];
    accR = __builtin_amdgcn_wmma_f32_16x16x4_f32(false, a, false, bR,
                                                 (short)0, accR, false, false);
    accI = __builtin_amdgcn_wmma_f32_16x16x4_f32(false, a, false, bI,
                                                 (short)0, accI, false, false);
  }

  // ---- Store D-matrix: VGPR j holds M=j (lanes 0-15) / M=j+8 (lanes 16-31) ----
  const int tg = t0 + tsub + nm;
#pragma unroll
  for (int j = 0; j < 8; ++j) {
    const int dg = d0 + j + half * 8;
    const size_t baseR = ((size_t)(b * 2 * D_ + dg)) * T_ + tg;
    const size_t baseI = ((size_t)(b * 2 * D_ + D_ + dg)) * T_ + tg;
    out[baseR] = accR[j];
    out[baseI] = accI[j];
  }
}

extern "C" void kernel_launch(void* const* d_in, const int* in_sizes, int n_in,
                              void* d_out, int out_size, void* d_ws, size_t ws_size,
                              hipStream_t stream) {
  (void)in_sizes; (void)n_in; (void)out_size; (void)d_ws; (void)ws_size;
  const float* xr = (const float*)d_in[0];
  const float* xi = (const float*)d_in[1];
  const float* fr = (const float*)d_in[2];
  const float* fi = (const float*)d_in[3];
  // d_in[4] = L (2), d_in[5] = I (1): fixed by the reference; tiling hardcodes them.
  float* out = (float*)d_out;

  dim3 grid(T_ / TT, D_ / DT, B_);   // 16 x 16 x 8 = 2048 workgroups
  deepfilter_kernel<<<grid, 256, 0, stream>>>(xr, xi, fr, fi, out);
}